// AdaptiveRouter_14851996909958
// MI455X (gfx1250) — compile-verified
//
#include <hip/hip_runtime.h>
#include <math.h>

// ---------------------------------------------------------------------------
// AdaptiveRouter for MI455X (gfx1250, wave32)
//   Kernel 1: tiny TTHA head (B=1) -> routing_biases, uncertainties  (1 block)
//   Kernel 2: x @ W_gate via V_WMMA_F32_16X16X4_F32 + bias + top-2 + softmax
// ---------------------------------------------------------------------------

typedef float v2f __attribute__((ext_vector_type(2)));
typedef float v8f __attribute__((ext_vector_type(8)));

#define NTOK 32768
#define DDIM 1024
#define NEXP 64
#define HH   256
#define HDIM 32

// output layout (floats): [ weights N*2 | idx N*2 | rb 64 | unc 64 ]
#define OFF_IDX (NTOK * 2)
#define OFF_RB  (NTOK * 4)
#define OFF_UN  (NTOK * 4 + 64)

__device__ __forceinline__ float gelu_exact(float x) {
  return 0.5f * x * (1.0f + erff(x * 0.70710678118654752f));
}

// LayerNorm across 256 values (one per thread) with block reductions.
__device__ float block_ln(float v, float g, float be, float* red, float* stats,
                          int t) {
  __syncthreads();
  red[t] = v;
  __syncthreads();
  if (t == 0) {
    float s = 0.f;
    for (int i = 0; i < HH; ++i) s += red[i];
    stats[0] = s * (1.0f / HH);
  }
  __syncthreads();
  float d = v - stats[0];
  __syncthreads();
  red[t] = d * d;
  __syncthreads();
  if (t == 0) {
    float s = 0.f;
    for (int i = 0; i < HH; ++i) s += red[i];
    stats[1] = rsqrtf(s * (1.0f / HH) + 1e-5f);
  }
  __syncthreads();
  return d * stats[1] * g + be;
}

__global__ __launch_bounds__(256) void ttha_kernel(
    const float* __restrict__ cost, const float* __restrict__ hw,
    const float* __restrict__ Wc, const float* __restrict__ bc,
    const float* __restrict__ gc, const float* __restrict__ bec,
    const float* __restrict__ Wh, const float* __restrict__ bh,
    const float* __restrict__ gh, const float* __restrict__ beh,
    const float* __restrict__ Wqkv, const float* __restrict__ bqkv,
    const float* __restrict__ Wo, const float* __restrict__ bo,
    const float* __restrict__ Wf, const float* __restrict__ bf,
    const float* __restrict__ gf, const float* __restrict__ bef,
    const float* __restrict__ Wo1, const float* __restrict__ bo1,
    const float* __restrict__ Wo2, const float* __restrict__ bo2,
    const float* __restrict__ Wu1, const float* __restrict__ bu1,
    const float* __restrict__ Wu2, const float* __restrict__ bu2,
    float* __restrict__ out, float* __restrict__ ws) {
  __shared__ float comb[3][HH];
  __shared__ float qkv[3][3 * HH];
  __shared__ float ctx[3][HH];
  __shared__ float mctx[HH];
  __shared__ float am[HH];
  __shared__ float fused[HH];
  __shared__ float o1[HH / 2];
  __shared__ float u1[NEXP];
  __shared__ float red[HH];
  __shared__ float stats[2];

  const int t = threadIdx.x;

  // ---- cost embed: gelu(ln(cost @ Wc + bc)) ----
  {
    float v = bc[t];
    for (int i = 0; i < 6 * NEXP; ++i) v += cost[i] * Wc[i * HH + t];
    comb[0][t] = gelu_exact(block_ln(v, gc[t], bec[t], red, stats, t));
  }
  // ---- hw embed ----
  {
    float v = bh[t];
    for (int i = 0; i < 8; ++i) v += hw[i] * Wh[i * HH + t];
    comb[1][t] = gelu_exact(block_ln(v, gh[t], beh[t], red, stats, t));
  }
  comb[2][t] = 0.0f;  // temporal_features=None path
  __syncthreads();

  // ---- qkv = comb @ Wqkv + bqkv   [3,768] ----
  for (int rep = 0; rep < 3; ++rep) {
    int j = t + rep * HH;  // 0..767
    for (int tkn = 0; tkn < 3; ++tkn) {
      float s = bqkv[j];
      if (tkn < 2)
        for (int i = 0; i < HH; ++i) s += comb[tkn][i] * Wqkv[i * (3 * HH) + j];
      qkv[tkn][j] = s;
    }
  }
  __syncthreads();

  // ---- 8-head attention over T=3 ----
  for (int rep = 0; rep < 3; ++rep) {
    int flat = t + rep * HH;  // (tkn, h, d)
    int tkn = flat >> 8;
    int rem = flat & 255;
    int h = rem >> 5;
    int d = rem & 31;
    float sc[3], mx = -1e30f;
    for (int kt = 0; kt < 3; ++kt) {
      float s = 0.f;
      for (int i = 0; i < HDIM; ++i)
        s += qkv[tkn][h * HDIM + i] * qkv[kt][HH + h * HDIM + i];
      s *= 0.17677669529663687f;  // 1/sqrt(32)
      sc[kt] = s;
      mx = fmaxf(mx, s);
    }
    float e0 = expf(sc[0] - mx), e1 = expf(sc[1] - mx), e2 = expf(sc[2] - mx);
    float inv = 1.0f / (e0 + e1 + e2);
    ctx[tkn][rem] = (e0 * qkv[0][2 * HH + h * HDIM + d] +
                     e1 * qkv[1][2 * HH + h * HDIM + d] +
                     e2 * qkv[2][2 * HH + h * HDIM + d]) *
                    inv;
  }
  __syncthreads();

  mctx[t] = (ctx[0][t] + ctx[1][t] + ctx[2][t]) * (1.0f / 3.0f);
  __syncthreads();

  // attended.mean = mean(ctx) @ Wo + bo
  {
    float s = bo[t];
    for (int i = 0; i < HH; ++i) s += mctx[i] * Wo[i * HH + t];
    am[t] = s;
  }
  __syncthreads();

  // fused = gelu(ln(am @ Wf + bf))
  {
    float s = bf[t];
    for (int i = 0; i < HH; ++i) s += am[i] * Wf[i * HH + t];
    fused[t] = gelu_exact(block_ln(s, gf[t], bef[t], red, stats, t));
  }
  __syncthreads();

  if (t < 128) {
    float s = bo1[t];
    for (int i = 0; i < HH; ++i) s += fused[i] * Wo1[i * 128 + t];
    o1[t] = gelu_exact(s);
  }
  if (t < 64) {
    float s = bu1[t];
    for (int i = 0; i < HH; ++i) s += fused[i] * Wu1[i * 64 + t];
    u1[t] = gelu_exact(s);
  }
  __syncthreads();

  if (t < 64) {
    float s = bo2[t];
    for (int i = 0; i < 128; ++i) s += o1[i] * Wo2[i * 64 + t];
    float rb = tanhf(s);
    ws[t] = rb;              // consumed by the gating kernel
    out[OFF_RB + t] = rb;
    float su = bu2[t];
    for (int i = 0; i < 64; ++i) su += u1[i] * Wu2[i * 64 + t];
    out[OFF_UN + t] = fmaxf(su, 0.f) + log1pf(expf(-fabsf(su)));  // softplus
  }
}

// ---------------------------------------------------------------------------
// Gating GEMM: 256 blocks x 256 threads (8 waves). Wave w handles rows
// [blk*128 + w*16, +16), all 64 experts = four 16x16 f32 WMMA tiles.
// W_gate is pre-swizzled into LDS in exact B-fragment layout:
//   entry[(s*4 + t)*32 + lane] = { W[k0][n], W[k0+1][n] },
//   k0 = 4s + 2*(lane>=16), n = 16t + (lane&15)
// A fragment layout (16x4 f32): lane<16 -> M=lane, K0/K1; lane>=16 -> K2/K3.
// ---------------------------------------------------------------------------
__global__ __launch_bounds__(256) void gate_kernel(
    const float* __restrict__ x, const float* __restrict__ Wg,
    const float* __restrict__ bg, const float* __restrict__ rbias,
    float* __restrict__ out) {
  extern __shared__ float smem[];  // 256 KB dynamic LDS
  const int tid = threadIdx.x;
  const int wave = tid >> 5;
  const int lane = tid & 31;
  const int l15 = lane & 15;
  const int hi = lane >> 4;

  // stage W fragments (once per block; W is L2-resident at 256 KB)
  v2f* bw = (v2f*)smem;  // 256 steps * 4 tiles * 32 lanes
  for (int e = tid; e < 256 * 4 * 32; e += 256) {
    int l = e & 31;
    int tt = (e >> 5) & 3;
    int s = e >> 7;
    int k0 = (s << 2) + ((l >> 4) << 1);
    int n = tt * 16 + (l & 15);
    v2f w;
    w.x = Wg[k0 * NEXP + n];
    w.y = Wg[(k0 + 1) * NEXP + n];
    bw[e] = w;
  }
  __syncthreads();

  const int rowBase = blockIdx.x * 128 + wave * 16;
  const float* ap = x + (size_t)(rowBase + l15) * DDIM + (hi << 1);

  v8f c0 = {}, c1 = {}, c2 = {}, c3 = {};
  for (int s = 0; s < 256; ++s) {
    v2f a = __builtin_nontemporal_load((const v2f*)(ap + s * 4));
    const v2f* bs = bw + s * 128 + lane;
    v2f b0 = bs[0];
    v2f b1 = bs[32];
    v2f b2 = bs[64];
    v2f b3 = bs[96];
    c0 = __builtin_amdgcn_wmma_f32_16x16x4_f32(false, a, false, b0, (short)0,
                                               c0, false, false);
    c1 = __builtin_amdgcn_wmma_f32_16x16x4_f32(false, a, false, b1, (short)0,
                                               c1, false, false);
    c2 = __builtin_amdgcn_wmma_f32_16x16x4_f32(false, a, false, b2, (short)0,
                                               c2, false, false);
    c3 = __builtin_amdgcn_wmma_f32_16x16x4_f32(false, a, false, b3, (short)0,
                                               c3, false, false);
  }
  __syncthreads();  // everyone done reading bw; reuse LDS for top-k staging

  // fold b_gate + routing_biases while spilling logits to LDS
  const float rb0 = bg[l15] + rbias[l15];
  const float rb1 = bg[16 + l15] + rbias[16 + l15];
  const float rb2 = bg[32 + l15] + rbias[32 + l15];
  const float rb3 = bg[48 + l15] + rbias[48 + l15];

  float* stage = smem + wave * (16 * 68);  // stride 68: bank-conflict-free
#pragma unroll
  for (int j = 0; j < 8; ++j) {
    int r = hi ? (j + 8) : j;  // C layout: vgpr j = rows j / j+8
    stage[r * 68 + 0 + l15] = c0[j] + rb0;
    stage[r * 68 + 16 + l15] = c1[j] + rb1;
    stage[r * 68 + 32 + l15] = c2[j] + rb2;
    stage[r * 68 + 48 + l15] = c3[j] + rb3;
  }
  __syncthreads();

  if (lane < 16) {
    const float* rowv = stage + lane * 68;
    float v1 = -1e30f, v2 = -1e30f;
    int i1 = 0, i2 = 0;
    for (int e = 0; e < 64; ++e) {  // strict '>' => lowest-index tie-break
      float v = rowv[e];
      if (v > v1) {
        v2 = v1; i2 = i1; v1 = v; i1 = e;
      } else if (v > v2) {
        v2 = v; i2 = e;
      }
    }
    float ex = expf(v2 - v1);
    float inv = 1.0f / (1.0f + ex);
    int gr = rowBase + lane;
    out[gr * 2 + 0] = inv;       // softmax over top-2
    out[gr * 2 + 1] = ex * inv;
    out[OFF_IDX + gr * 2 + 0] = (float)i1;
    out[OFF_IDX + gr * 2 + 1] = (float)i2;
  }
}

extern "C" void kernel_launch(void* const* d_in, const int* in_sizes, int n_in,
                              void* d_out, int out_size, void* d_ws,
                              size_t ws_size, hipStream_t stream) {
  const float* x    = (const float*)d_in[0];
  const float* cost = (const float*)d_in[1];
  const float* hw   = (const float*)d_in[2];
  const float* Wg   = (const float*)d_in[3];
  const float* bg   = (const float*)d_in[4];
  const float* Wc   = (const float*)d_in[5];
  const float* bc   = (const float*)d_in[6];
  const float* gc   = (const float*)d_in[7];
  const float* bec  = (const float*)d_in[8];
  const float* Wh   = (const float*)d_in[9];
  const float* bh   = (const float*)d_in[10];
  const float* gh   = (const float*)d_in[11];
  const float* beh  = (const float*)d_in[12];
  const float* Wqkv = (const float*)d_in[13];
  const float* bqkv = (const float*)d_in[14];
  const float* Wo   = (const float*)d_in[15];
  const float* bo   = (const float*)d_in[16];
  const float* Wf   = (const float*)d_in[17];
  const float* bf   = (const float*)d_in[18];
  const float* gf   = (const float*)d_in[19];
  const float* bef  = (const float*)d_in[20];
  const float* Wo1  = (const float*)d_in[21];
  const float* bo1  = (const float*)d_in[22];
  const float* Wo2  = (const float*)d_in[23];
  const float* bo2  = (const float*)d_in[24];
  const float* Wu1  = (const float*)d_in[25];
  const float* bu1  = (const float*)d_in[26];
  const float* Wu2  = (const float*)d_in[27];
  const float* bu2  = (const float*)d_in[28];

  float* out = (float*)d_out;
  float* ws  = (float*)d_ws;

  (void)hipFuncSetAttribute((const void*)gate_kernel,
                            hipFuncAttributeMaxDynamicSharedMemorySize,
                            256 * 1024);

  ttha_kernel<<<1, 256, 0, stream>>>(cost, hw, Wc, bc, gc, bec, Wh, bh, gh,
                                     beh, Wqkv, bqkv, Wo, bo, Wf, bf, gf, bef,
                                     Wo1, bo1, Wo2, bo2, Wu1, bu1, Wu2, bu2,
                                     out, ws);

  gate_kernel<<<NTOK / 128, 256, 256 * 1024, stream>>>(x, Wg, bg, ws, out);
}